// AttnDecoderRNN_69114613730222
// MI455X (gfx1250) — compile-verified
//
#include <hip/hip_runtime.h>
#include <math.h>

typedef __attribute__((ext_vector_type(2))) float v2f;
typedef __attribute__((ext_vector_type(8))) float v8f;

// Problem dims
#define VOCAB 50257
#define DIM   512
#define H     1024
#define S     2048
#define L     16
#define TWO_H 2048
#define KCOMB (DIM + TWO_H + L)   // 2576
#define KOUT  (H + L + DIM)       // 1552

// Workspace float offsets
#define WS_V      0        // 2048
#define WS_SLOG   2048     // 2048
#define WS_SCORES 4096     // 2048
#define WS_X      6144     // 2576
#define WS_H1     8720     // 1024
#define WS_H2     9744     // 1024
#define WS_H3     10768    // 1024
#define WS_GI     11792    // 3072
#define WS_GH     14864    // 3072
#define WS_XO     17936    // 1552
#define WS_PMAX   19488    // 64
#define WS_PSUM   19552    // 64
#define WS_SC     19616    // 2

// ---------------------------------------------------------------------------
// WMMA matvec: y[r] = act( W[r,:] . x + bias[r] ), W row-major [nrows x K],
// K % 4 == 0. One wave computes 16 rows via V_WMMA_F32_16X16X4_F32 chain,
// using only column N=0 of the B/D tiles (memory-bound, so free).
// ---------------------------------------------------------------------------
__global__ __launch_bounds__(256) void k_wmma_mv(
    const float* __restrict__ W, const float* __restrict__ x,
    const float* __restrict__ bias, float* __restrict__ y,
    int nrows, int K, int relu_act) {
  __shared__ float sx[KCOMB];  // max K = 2576
  for (int i = threadIdx.x; i < K; i += blockDim.x) sx[i] = x[i];
  __syncthreads();

  const int lane = threadIdx.x & 31;
  const int wave = threadIdx.x >> 5;
  const int tile = blockIdx.x * (blockDim.x >> 5) + wave;
  const int row0 = tile * 16;
  if (row0 >= nrows) return;                 // wave-uniform exit

  const int rit = lane & 15;                 // row within 16-row tile (A: M)
  const int kh  = lane >> 4;                 // K-pair select (A: K=0/1 vs 2/3)
  int r = row0 + rit;
  if (r >= nrows) r = nrows - 1;             // clamp loads; stores masked below
  const float* wrow = W + (size_t)r * K;

  v8f acc = {};
  for (int c = 0; c < K; c += 4) {
    // A 16x4 f32 tile (ISA layout): lane m holds K={0,1} (kh=0) or K={2,3} (kh=1)
    v2f a = *(const v2f*)(wrow + c + 2 * kh);
    // B 4x16: column N=0 only -> lane 0 holds B[0..1][0], lane 16 holds B[2..3][0]
    float b0 = sx[c + 2 * kh];
    float b1 = sx[c + 2 * kh + 1];
    v2f b;
    b.x = (rit == 0) ? b0 : 0.0f;
    b.y = (rit == 0) ? b1 : 0.0f;
    acc = __builtin_amdgcn_wmma_f32_16x16x4_f32(
        /*neg_a=*/false, a, /*neg_b=*/false, b,
        /*c_mod=*/(short)0, acc, /*reuse_a=*/false, /*reuse_b=*/false);
  }

  // D column N=0: lane 0 VGPR g -> row g; lane 16 VGPR g -> row 8+g
  if (rit == 0) {
    const int rb = row0 + 8 * kh;
#pragma unroll
    for (int g = 0; g < 8; ++g) {
      int rr = rb + g;
      if (rr < nrows) {
        float val = acc[g];
        if (bias) val += bias[rr];
        if (relu_act) val = fmaxf(val, 0.0f);
        y[rr] = val;
      }
    }
  }
}

// ---------------------------------------------------------------------------
// Column matvec (transposed weight access): y[j] = sum_i W[i*nout+j] * x[i]
// Coalesced across j; x staged in LDS (broadcast).
// ---------------------------------------------------------------------------
__global__ __launch_bounds__(256) void k_colmv(
    const float* __restrict__ W, const float* __restrict__ x,
    float* __restrict__ y, int nin, int nout) {
  __shared__ float sxv[TWO_H];  // max nin = 2048
  for (int i = threadIdx.x; i < nin; i += blockDim.x) sxv[i] = x[i];
  __syncthreads();
  int j = blockIdx.x * blockDim.x + threadIdx.x;
  if (j >= nout) return;
  float acc = 0.0f;
  for (int i = 0; i < nin; ++i)
    acc = fmaf(W[(size_t)i * nout + j], sxv[i], acc);
  y[j] = acc;
}

// ---------------------------------------------------------------------------
// Softmax over S=2048 (single block). Writes scores (ws) and attn_weights out.
// ---------------------------------------------------------------------------
__global__ __launch_bounds__(256) void k_softmax_s(
    const float* __restrict__ lg, float* __restrict__ scores,
    float* __restrict__ attn) {
  __shared__ float red[256];
  const int t = threadIdx.x;
  float m = -3.402823466e38f;
  for (int i = t; i < S; i += 256) m = fmaxf(m, lg[i]);
  red[t] = m; __syncthreads();
  for (int s = 128; s > 0; s >>= 1) {
    if (t < s) red[t] = fmaxf(red[t], red[t + s]);
    __syncthreads();
  }
  const float mx = red[0]; __syncthreads();
  float sum = 0.0f;
  for (int i = t; i < S; i += 256) sum += expf(lg[i] - mx);
  red[t] = sum; __syncthreads();
  for (int s = 128; s > 0; s >>= 1) {
    if (t < s) red[t] += red[t + s];
    __syncthreads();
  }
  const float inv = 1.0f / red[0];
  for (int i = t; i < S; i += 256) {
    float v = expf(lg[i] - mx) * inv;
    scores[i] = v;
    attn[i] = v;
  }
}

// Gather embedding row + scatter nlg into both concat vectors.
__global__ __launch_bounds__(256) void k_gather(
    const int* __restrict__ token, const float* __restrict__ emb,
    const float* __restrict__ nlg, float* __restrict__ x,
    float* __restrict__ xo) {
  const int t = token[0];
  const float* e = emb + (size_t)t * DIM;
  int i = blockIdx.x * blockDim.x + threadIdx.x;
  if (i < DIM) { x[i] = e[i]; xo[H + L + i] = e[i]; }
  if (i < L)   { x[DIM + TWO_H + i] = nlg[i]; xo[H + i] = nlg[i]; }
}

__device__ __forceinline__ float sigmoidf(float v) {
  return 1.0f / (1.0f + expf(-v));
}

// GRU combine (PyTorch r,z,n ordering). Writes h_new to output and to xo[0:H].
__global__ __launch_bounds__(256) void k_gru(
    const float* __restrict__ gi, const float* __restrict__ gh,
    const float* __restrict__ hprev, float* __restrict__ hnew,
    float* __restrict__ xo) {
  int j = blockIdx.x * blockDim.x + threadIdx.x;
  if (j >= H) return;
  float r = sigmoidf(gi[j] + gh[j]);
  float z = sigmoidf(gi[H + j] + gh[H + j]);
  float n = tanhf(gi[2 * H + j] + r * gh[2 * H + j]);
  float h = (1.0f - z) * n + z * hprev[j];
  hnew[j] = h;
  xo[j] = h;
}

// ---- deterministic two-stage log_softmax over VOCAB ----
__global__ __launch_bounds__(256) void k_blk_max(
    const float* __restrict__ l, int n, float* __restrict__ pmax) {
  __shared__ float red[256];
  const int t = threadIdx.x;
  float m = -3.402823466e38f;
  for (int i = blockIdx.x * 256 + t; i < n; i += gridDim.x * 256)
    m = fmaxf(m, l[i]);
  red[t] = m; __syncthreads();
  for (int s = 128; s > 0; s >>= 1) {
    if (t < s) red[t] = fmaxf(red[t], red[t + s]);
    __syncthreads();
  }
  if (t == 0) pmax[blockIdx.x] = red[0];
}

__global__ void k_final_max(const float* __restrict__ pmax, int nb,
                            float* __restrict__ sc) {
  if (threadIdx.x == 0 && blockIdx.x == 0) {
    float m = -3.402823466e38f;
    for (int i = 0; i < nb; ++i) m = fmaxf(m, pmax[i]);
    sc[0] = m;
  }
}

__global__ __launch_bounds__(256) void k_blk_sum(
    const float* __restrict__ l, int n, const float* __restrict__ sc,
    float* __restrict__ psum) {
  __shared__ float red[256];
  const int t = threadIdx.x;
  const float mx = sc[0];
  float s = 0.0f;
  for (int i = blockIdx.x * 256 + t; i < n; i += gridDim.x * 256)
    s += expf(l[i] - mx);
  red[t] = s; __syncthreads();
  for (int k = 128; k > 0; k >>= 1) {
    if (t < k) red[t] += red[t + k];
    __syncthreads();
  }
  if (t == 0) psum[blockIdx.x] = red[0];
}

__global__ void k_final_sum(const float* __restrict__ psum, int nb,
                            float* __restrict__ sc) {
  if (threadIdx.x == 0 && blockIdx.x == 0) {
    float s = 0.0f;
    for (int i = 0; i < nb; ++i) s += psum[i];
    sc[1] = logf(s);
  }
}

__global__ __launch_bounds__(256) void k_ls_write(
    float* __restrict__ l, int n, const float* __restrict__ sc) {
  int i = blockIdx.x * blockDim.x + threadIdx.x;
  if (i < n) l[i] = l[i] - sc[0] - sc[1];
}

// ---------------------------------------------------------------------------
extern "C" void kernel_launch(void* const* d_in, const int* in_sizes, int n_in,
                              void* d_out, int out_size, void* d_ws, size_t ws_size,
                              hipStream_t stream) {
  const int*   token  = (const int*)d_in[0];
  const float* hidden = (const float*)d_in[1];
  const float* eo     = (const float*)d_in[2];
  const float* nlg    = (const float*)d_in[3];
  const float* emb    = (const float*)d_in[4];
  const float* attn_W = (const float*)d_in[5];
  // d_in[6] = attn_b: contributes a constant to all attention logits -> cancels in softmax
  const float* comb_W = (const float*)d_in[7];
  const float* comb_b = (const float*)d_in[8];
  const float* fc1_W  = (const float*)d_in[9];
  const float* fc1_b  = (const float*)d_in[10];
  const float* fc2_W  = (const float*)d_in[11];
  const float* fc2_b  = (const float*)d_in[12];
  const float* Wih    = (const float*)d_in[13];
  const float* Whh    = (const float*)d_in[14];
  const float* bih    = (const float*)d_in[15];
  const float* bhh    = (const float*)d_in[16];
  const float* out_W  = (const float*)d_in[17];
  const float* out_b  = (const float*)d_in[18];

  float* out  = (float*)d_out;             // [VOCAB] log-probs
  float* hnew = out + VOCAB;               // [H]
  float* attn = out + VOCAB + H;           // [S]
  float* ws   = (float*)d_ws;

  const int B = 256;                       // 8 waves/block, 128 rows/block for wmma_mv
  auto mvGrid = [](int rows) { return (rows + 127) / 128; };

  // 1) v = attn_W^T h   (column matvec: [H x 2H] -> 2H)
  k_colmv<<<(TWO_H + B - 1) / B, B, 0, stream>>>(attn_W, hidden, ws + WS_V, H, TWO_H);
  // 2) slogit[s] = eo[s] . v   (WMMA matvec, rows=S, K=2H)
  k_wmma_mv<<<mvGrid(S), B, 0, stream>>>(eo, ws + WS_V, nullptr, ws + WS_SLOG, S, TWO_H, 0);
  // 3) softmax -> scores + attn_weights output
  k_softmax_s<<<1, B, 0, stream>>>(ws + WS_SLOG, ws + WS_SCORES, attn);
  // 4) context = scores^T eo  -> x[512:2560]
  k_colmv<<<(TWO_H + B - 1) / B, B, 0, stream>>>(eo, ws + WS_SCORES, ws + WS_X + DIM, S, TWO_H);
  // 5) emb + nlg into x and xo
  k_gather<<<(DIM + B - 1) / B, B, 0, stream>>>(token, emb, nlg, ws + WS_X, ws + WS_XO);
  // 6) h1 = relu(comb_W @ x + comb_b)
  k_wmma_mv<<<mvGrid(H), B, 0, stream>>>(comb_W, ws + WS_X, comb_b, ws + WS_H1, H, KCOMB, 1);
  // 7) h2 = relu(fc1_W @ h1 + fc1_b)
  k_wmma_mv<<<mvGrid(H), B, 0, stream>>>(fc1_W, ws + WS_H1, fc1_b, ws + WS_H2, H, H, 1);
  // 8) h3 = relu(fc2_W @ h2 + fc2_b)
  k_wmma_mv<<<mvGrid(H), B, 0, stream>>>(fc2_W, ws + WS_H2, fc2_b, ws + WS_H3, H, H, 1);
  // 9) gi = Wih @ h3 + bih ; gh = Whh @ h_prev + bhh
  k_wmma_mv<<<mvGrid(3 * H), B, 0, stream>>>(Wih, ws + WS_H3, bih, ws + WS_GI, 3 * H, H, 0);
  k_wmma_mv<<<mvGrid(3 * H), B, 0, stream>>>(Whh, hidden, bhh, ws + WS_GH, 3 * H, H, 0);
  // 10) GRU combine -> h_new (output) and xo[0:H]
  k_gru<<<(H + B - 1) / B, B, 0, stream>>>(ws + WS_GI, ws + WS_GH, hidden, hnew, ws + WS_XO);
  // 11) logits = out_W @ xo + out_b   (the 312 MB streaming pass)
  k_wmma_mv<<<mvGrid(VOCAB), B, 0, stream>>>(out_W, ws + WS_XO, out_b, out, VOCAB, KOUT, 0);
  // 12) log_softmax over VOCAB (deterministic two-stage reductions)
  k_blk_max<<<64, B, 0, stream>>>(out, VOCAB, ws + WS_PMAX);
  k_final_max<<<1, 64, 0, stream>>>(ws + WS_PMAX, 64, ws + WS_SC);
  k_blk_sum<<<64, B, 0, stream>>>(out, VOCAB, ws + WS_SC, ws + WS_PSUM);
  k_final_sum<<<1, 64, 0, stream>>>(ws + WS_PSUM, 64, ws + WS_SC);
  k_ls_write<<<(VOCAB + B - 1) / B, B, 0, stream>>>(out, VOCAB, ws + WS_SC);
}